// BasicConvolutionBlockdown_25898652795312
// MI455X (gfx1250) — compile-verified
//
#include <hip/hip_runtime.h>

// ---------------------------------------------------------------------------
// Sparse conv (gather-GEMM-scatter, scatter-first form) + BatchNorm, fp32.
//   seg = out_idx*8 + offset_id
//   S[M*8, 32]  = segment_sum(feats)                (atomic scatter)
//   out[M, 64]  = S.reshape(M,256) @ W.reshape(256,64)   (WMMA f32 16x16x4)
//   y = (out - mean) * rsqrt(var + eps) * gamma + beta   (batch stats over M)
// ---------------------------------------------------------------------------

typedef __attribute__((ext_vector_type(2))) float v2f;
typedef __attribute__((ext_vector_type(8))) float v8f;

#define INC    32
#define OUTC   64
#define KVOL   8
#define KDIM   256          // KVOL * INC
#define BN_EPS 1e-5f

// ---- kernel 1: zero the scatter workspace + BN stats ----------------------
__global__ void spconv_zero_ws(float* __restrict__ ws, long long nelem4) {
    long long i = (long long)blockIdx.x * blockDim.x + threadIdx.x;
    long long stride = (long long)gridDim.x * blockDim.x;
    float4* p = (float4*)ws;
    for (long long j = i; j < nelem4; j += stride)
        p[j] = make_float4(0.f, 0.f, 0.f, 0.f);
}

// ---- kernel 2: scatter feats into (out_voxel, kernel_offset) buckets ------
// one wave (32 lanes) per input voxel; lane = channel. Coalesced 128B
// loads and 32 consecutive f32 atomics per voxel (global_atomic_add_f32).
__global__ void spconv_scatter(const float* __restrict__ feats,
                               const int*   __restrict__ out_idx,
                               const int*   __restrict__ offset_id,
                               float*       __restrict__ S,
                               int nvox) {
    int t   = blockIdx.x * blockDim.x + threadIdx.x;
    int vox = t >> 5;
    int c   = t & 31;
    if (vox >= nvox) return;
    int seg = out_idx[vox] * KVOL + offset_id[vox];
    unsafeAtomicAdd(&S[(long long)seg * INC + c],
                    feats[(long long)vox * INC + c]);
}

// ---- kernel 3: GEMM out = S[M,256] @ W[256,64] via V_WMMA_F32_16X16X4_F32 -
// 256 threads = 8 waves; each wave owns a 32x64 output tile (2 m-groups x
// 4 n-tiles = 8 accumulators). Weights staged in LDS *K-pair interleaved*
// (wint[p*128 + n*2 + e] = W[2p+e][n]) so each lane's B operand
// (W[k+kb][col], W[k+kb+1][col]) is one aligned ds_load_b64 — no packing
// moves. Per-column BN sums reduced via shfl_xor(16) + f32 global atomics.
__launch_bounds__(256)
__global__ void spconv_gemm_stats(const float* __restrict__ S,
                                  const float* __restrict__ weight,
                                  float*       __restrict__ out,
                                  float*       __restrict__ stats,  // [0:64)=sum, [64:128)=sumsq
                                  int mtotal) {
    __shared__ float wint[(KDIM / 2) * 2 * OUTC];   // 128 pairs * 128 = 64 KB

    // interleaved staging: wint[p*128 + n*2 + e] = W[(2p+e)*64 + n]
    for (int i = threadIdx.x; i < (KDIM / 2) * OUTC; i += blockDim.x) {
        int p = i >> 6;      // K-pair index 0..127
        int n = i & 63;      // output column
        wint[p * 128 + n * 2 + 0] = weight[(2 * p)     * OUTC + n];
        wint[p * 128 + n * 2 + 1] = weight[(2 * p + 1) * OUTC + n];
    }
    __syncthreads();

    const int wave = threadIdx.x >> 5;
    const int lane = threadIdx.x & 31;
    const int r    = lane & 15;     // M-row (A) / N-col (B,C,D) within tile
    const int kbh  = lane >> 4;     // which K sub-pair this half-wave owns

    const int row_base = (blockIdx.x * 8 + wave) * 32;
    if (row_base >= mtotal) return;   // wave-uniform, after the only barrier

    // A operands: lane holds A[row][k + 2*kbh + {0,1}]; clamp tail rows
    const int ar0 = min(row_base + r,      mtotal - 1);
    const int ar1 = min(row_base + 16 + r, mtotal - 1);
    const float* __restrict__ A0 = S + (long long)ar0 * KDIM + 2 * kbh;
    const float* __restrict__ A1 = S + (long long)ar1 * KDIM + 2 * kbh;
    const float* __restrict__ B  = wint + kbh * 128 + r * 2;

    v8f a00 = {}, a01 = {}, a02 = {}, a03 = {};   // rows [row_base, +16)
    v8f a10 = {}, a11 = {}, a12 = {}, a13 = {};   // rows [row_base+16, +32)

#pragma unroll 4
    for (int k = 0; k < KDIM; k += 4) {
        float2 f0 = *(const float2*)(A0 + k);
        float2 f1 = *(const float2*)(A1 + k);
        v2f va0; va0.x = f0.x; va0.y = f0.y;
        v2f va1; va1.x = f1.x; va1.y = f1.y;

        const float* bp = B + (k >> 1) * 128;     // advance one K-pair per 2 k
        float2 g0 = *(const float2*)(bp);
        float2 g1 = *(const float2*)(bp + 32);
        float2 g2 = *(const float2*)(bp + 64);
        float2 g3 = *(const float2*)(bp + 96);
        v2f b0; b0.x = g0.x; b0.y = g0.y;
        v2f b1; b1.x = g1.x; b1.y = g1.y;
        v2f b2; b2.x = g2.x; b2.y = g2.y;
        v2f b3; b3.x = g3.x; b3.y = g3.y;

        a00 = __builtin_amdgcn_wmma_f32_16x16x4_f32(false, va0, false, b0, (short)0, a00, false, false);
        a10 = __builtin_amdgcn_wmma_f32_16x16x4_f32(false, va1, false, b0, (short)0, a10, false, false);
        a01 = __builtin_amdgcn_wmma_f32_16x16x4_f32(false, va0, false, b1, (short)0, a01, false, false);
        a11 = __builtin_amdgcn_wmma_f32_16x16x4_f32(false, va1, false, b1, (short)0, a11, false, false);
        a02 = __builtin_amdgcn_wmma_f32_16x16x4_f32(false, va0, false, b2, (short)0, a02, false, false);
        a12 = __builtin_amdgcn_wmma_f32_16x16x4_f32(false, va1, false, b2, (short)0, a12, false, false);
        a03 = __builtin_amdgcn_wmma_f32_16x16x4_f32(false, va0, false, b3, (short)0, a03, false, false);
        a13 = __builtin_amdgcn_wmma_f32_16x16x4_f32(false, va1, false, b3, (short)0, a13, false, false);
    }

    // D layout: VGPR i -> row (base + i + 8*(lane>=16)), col = nt*16 + r
    const int mrow0 = row_base + ((lane >> 4) << 3);

#define EMIT_NT(ACCL, ACCH, NT)                                                \
    {                                                                          \
        float ps = 0.f, pq = 0.f;                                              \
        _Pragma("unroll")                                                      \
        for (int i = 0; i < 8; ++i) {                                          \
            int m = mrow0 + i;                                                 \
            float v = (ACCL)[i];                                               \
            if (m < mtotal) {                                                  \
                out[(long long)m * OUTC + (NT) * 16 + r] = v;                  \
                ps += v; pq += v * v;                                          \
            }                                                                  \
        }                                                                      \
        _Pragma("unroll")                                                      \
        for (int i = 0; i < 8; ++i) {                                          \
            int m = mrow0 + 16 + i;                                            \
            float v = (ACCH)[i];                                               \
            if (m < mtotal) {                                                  \
                out[(long long)m * OUTC + (NT) * 16 + r] = v;                  \
                ps += v; pq += v * v;                                          \
            }                                                                  \
        }                                                                      \
        ps += __shfl_xor(ps, 16, 32);  /* lanes l, l^16 share a column */      \
        pq += __shfl_xor(pq, 16, 32);                                          \
        if (lane < 16) {                                                       \
            unsafeAtomicAdd(&stats[(NT) * 16 + r], ps);                        \
            unsafeAtomicAdd(&stats[64 + (NT) * 16 + r], pq);                   \
        }                                                                      \
    }

    EMIT_NT(a00, a10, 0)
    EMIT_NT(a01, a11, 1)
    EMIT_NT(a02, a12, 2)
    EMIT_NT(a03, a13, 3)
#undef EMIT_NT
}

// ---- kernel 4: apply batch norm in place ----------------------------------
__global__ void spconv_bn_apply(float* __restrict__ out,
                                const float* __restrict__ stats,
                                const float* __restrict__ gamma,
                                const float* __restrict__ beta,
                                int mtotal) {
    long long i4     = (long long)blockIdx.x * blockDim.x + threadIdx.x;
    long long total4 = ((long long)mtotal * OUTC) >> 2;
    if (i4 >= total4) return;
    const float inv_m = 1.0f / (float)mtotal;
    const int c0 = (int)((i4 << 2) & (OUTC - 1));   // OUTC % 4 == 0: no wrap
    float4 v = ((float4*)out)[i4];
    float x[4] = {v.x, v.y, v.z, v.w};
    float y[4];
#pragma unroll
    for (int j = 0; j < 4; ++j) {
        int c = c0 + j;
        float mean = stats[c] * inv_m;
        float var  = stats[64 + c] * inv_m - mean * mean;   // biased, matches BN
        y[j] = (x[j] - mean) * rsqrtf(var + BN_EPS) * gamma[c] + beta[c];
    }
    ((float4*)out)[i4] = make_float4(y[0], y[1], y[2], y[3]);
}

// ---------------------------------------------------------------------------
extern "C" void kernel_launch(void* const* d_in, const int* in_sizes, int n_in,
                              void* d_out, int out_size, void* d_ws, size_t ws_size,
                              hipStream_t stream) {
    const float* feats     = (const float*)d_in[0];   // [N, 32]
    const float* weight    = (const float*)d_in[1];   // [8, 32, 64] == [256, 64]
    const float* gamma     = (const float*)d_in[2];   // [64]
    const float* beta      = (const float*)d_in[3];   // [64]
    const int*   out_idx   = (const int*)d_in[4];     // [N]
    const int*   offset_id = (const int*)d_in[5];     // [N]
    // d_in[6] = num_out (device scalar); derive sizes host-side instead:
    const int nvox   = in_sizes[0] / INC;
    const int mtotal = out_size / OUTC;

    float* S     = (float*)d_ws;                          // [M*8, 32] = [M, 256]
    float* stats = S + (long long)mtotal * KDIM;          // 128 floats
    const long long zelem4 = ((long long)mtotal * KDIM + 128) >> 2;

    spconv_zero_ws<<<4096, 256, 0, stream>>>((float*)d_ws, zelem4);

    const long long sthreads = (long long)nvox * INC;
    const int sblocks = (int)((sthreads + 255) / 256);
    spconv_scatter<<<sblocks, 256, 0, stream>>>(feats, out_idx, offset_id, S, nvox);

    const int gblocks = (mtotal + 255) / 256;             // 8 waves * 32 rows
    spconv_gemm_stats<<<gblocks, 256, 0, stream>>>(S, weight, (float*)d_out, stats, mtotal);

    const long long total4 = ((long long)mtotal * OUTC) >> 2;
    const int bblocks = (int)((total4 + 255) / 256);
    spconv_bn_apply<<<bblocks, 256, 0, stream>>>((float*)d_out, stats, gamma, beta, mtotal);
}